// PermutohedralLatticeLayer_21242908246578
// MI455X (gfx1250) — compile-verified
//
#include <hip/hip_runtime.h>

// ---------------------------------------------------------------------------
// Permutohedral lattice encoding (D=3, L=16, F=2, T=2^18) fused with a 32x32
// dense layer done via V_WMMA_F32_16X16X4_F32 on gfx1250 (wave32).
// ---------------------------------------------------------------------------

#define DD          3
#define LL          16
#define FFEAT       2
#define LOG2T       18
#define TSIZE       (1 << LOG2T)
#define NPTS        262144
#define PTS_PER_BLK 128
#define THREADS     256
#define ENC_STRIDE  34          // 32 features + 2 floats pad (bank-conflict break)

typedef __attribute__((ext_vector_type(2))) float v2f;
typedef __attribute__((ext_vector_type(8))) float v8f;

// log2(FINEST/COARSEST)/(L-1) = log2(100)/15
#define LOG2_GROWTH 0.4429237465f
// INV_STD = 1/sqrt(d*(d+1)) for d=1..3
#define INV_STD0 0.70710678118f
#define INV_STD1 0.40824829046f
#define INV_STD2 0.28867513459f

__global__ __launch_bounds__(THREADS)
void permuto_wmma_kernel(const float* __restrict__ x,
                         const float* __restrict__ table,
                         const float* __restrict__ shifts,
                         const float* __restrict__ W,
                         const float* __restrict__ bvec,
                         float* __restrict__ out)
{
    __shared__ float lds_x[PTS_PER_BLK * 3];
    __shared__ float lds_enc[PTS_PER_BLK * ENC_STRIDE];

    const int tid     = threadIdx.x;
    const int base_pt = blockIdx.x * PTS_PER_BLK;

    // Warm the 4KB W matrix + bias into cache early (global_prefetch_b8).
    if (tid < 64) __builtin_prefetch(W + tid * 16, 0, 3);
    if (tid == 64) __builtin_prefetch(bvec, 0, 3);

    // ---- stage x tile into LDS (coalesced) -------------------------------
    for (int i = tid; i < PTS_PER_BLK * 3; i += THREADS)
        lds_x[i] = x[(size_t)base_pt * 3 + i];
    __syncthreads();

    // ---- phase 1: encoding. thread = (point-in-pass = tid&15, level = tid>>4)
    // Each half-wave shares one level -> gathers stay in one 2MB table slice.
    const int lvl  = tid >> 4;   // 0..15
    const int psub = tid & 15;   // 0..15

    const float scale = 10.0f * exp2f(LOG2_GROWTH * (float)lvl);
    const float sf0 = scale * INV_STD0;
    const float sf1 = scale * INV_STD1;
    const float sf2 = scale * INV_STD2;
    const float sh0 = shifts[lvl * 3 + 0];
    const float sh1 = shifts[lvl * 3 + 1];
    const float sh2 = shifts[lvl * 3 + 2];
    const float2* __restrict__ tbl = (const float2*)table + (size_t)lvl * TSIZE;

    #pragma unroll 2
    for (int pass = 0; pass < PTS_PER_BLK / 16; ++pass) {
        const int p = pass * 16 + psub;
        const float cf0 = (lds_x[p * 3 + 0] + sh0) * sf0;
        const float cf1 = (lds_x[p * 3 + 1] + sh1) * sf1;
        const float cf2 = (lds_x[p * 3 + 2] + sh2) * sf2;

        float e[4];
        e[0] =  cf0 + cf1 + cf2;
        e[1] = -cf0 + cf1 + cf2;
        e[2] = -2.0f * cf1 + cf2;
        e[3] = -3.0f * cf2;

        // round each elevated coord to nearest multiple of (d+1)=4
        float rem[4];
        float rsum = 0.0f;
        #pragma unroll
        for (int i = 0; i < 4; ++i) {
            const float v  = e[i] * 0.25f;
            const float up = ceilf(v)  * 4.0f;
            const float dn = floorf(v) * 4.0f;
            rem[i] = ((up - e[i]) < (e[i] - dn)) ? up : dn;
            rsum += rem[i];
        }
        const int s = (int)rintf(rsum * 0.25f);

        float diff[4];
        #pragma unroll
        for (int i = 0; i < 4; ++i) diff[i] = e[i] - rem[i];

        int rank[4];
        #pragma unroll
        for (int i = 0; i < 4; ++i) {
            int rk = s;
            #pragma unroll
            for (int j = 0; j < 4; ++j) {
                if (j == i) continue;
                rk += ((diff[j] > diff[i]) ||
                       ((diff[j] == diff[i]) && (j < i))) ? 1 : 0;
            }
            rank[i] = rk;
        }

        int remi[4];
        #pragma unroll
        for (int i = 0; i < 4; ++i) {
            remi[i] = (int)rem[i];
            if (rank[i] < 0)      { rank[i] += 4; remi[i] += 4; }
            else if (rank[i] > 3) { rank[i] -= 4; remi[i] -= 4; }
        }

        float delta[4];
        #pragma unroll
        for (int i = 0; i < 4; ++i) delta[i] = (e[i] - (float)remi[i]) * 0.25f;

        // barycentric weights: delta[i] -> +bary[3-rank[i]], -bary[4-rank[i]]
        float bary[5] = {0.f, 0.f, 0.f, 0.f, 0.f};
        #pragma unroll
        for (int i = 0; i < 4; ++i) {
            #pragma unroll
            for (int o = 0; o < 5; ++o) {
                float w = 0.0f;
                w += (rank[i] == (3 - o)) ? delta[i] : 0.0f;
                w -= (rank[i] == (4 - o)) ? delta[i] : 0.0f;
                bary[o] += w;
            }
        }
        bary[0] += 1.0f + bary[4];

        // hash the 4 simplex vertices and gather float2 features
        float acc0 = 0.0f, acc1 = 0.0f;
        #pragma unroll
        for (int r = 0; r < 4; ++r) {
            const unsigned k0 = (unsigned)(remi[0] + r - 4 * (rank[0] > 3 - r));
            const unsigned k1 = (unsigned)(remi[1] + r - 4 * (rank[1] > 3 - r));
            const unsigned k2 = (unsigned)(remi[2] + r - 4 * (rank[2] > 3 - r));
            unsigned h = k0;
            h ^= k1 * 2654435761u;
            h ^= k2 * 805459861u;
            const unsigned idx = h & (unsigned)(TSIZE - 1);
            const float2 f = tbl[idx];
            acc0 += bary[r] * f.x;
            acc1 += bary[r] * f.y;
        }

        *(v2f*)&lds_enc[p * ENC_STRIDE + 2 * lvl] = (v2f){acc0, acc1};
    }
    __syncthreads();

    // ---- phase 2: out[128,32] = enc[128,32] @ W^T + b via fp32 WMMA ------
    // wave w owns rows [w*16, w*16+16); two 16-col tiles; K=32 in 8 steps of 4.
    const int wave = tid >> 5;      // 0..7
    const int lane = tid & 31;
    const int ln   = lane & 15;     // N index / M index
    const int lh   = lane >> 4;     // K-pair select (A/B), M+8 select (C/D)

    const float* __restrict__ encrow = &lds_enc[(wave * 16 + ln) * ENC_STRIDE];

    #pragma unroll
    for (int nt = 0; nt < 2; ++nt) {
        v8f acc = {0.f, 0.f, 0.f, 0.f, 0.f, 0.f, 0.f, 0.f};
        const float* __restrict__ wrow = W + (nt * 16 + ln) * 32;  // B[k][n]=W[n][k]

        #pragma unroll
        for (int kk = 0; kk < 8; ++kk) {
            const int k = kk * 4 + 2 * lh;
            // A 16x4 f32: lane<16 -> K={k0,k0+1}, lane>=16 -> K={k0+2,k0+3}
            const v2f a = *(const v2f*)&encrow[k];
            // B 4x16 f32: same K striping, N = lane&15
            const v2f bm = *(const v2f*)&wrow[k];
            acc = __builtin_amdgcn_wmma_f32_16x16x4_f32(
                      false, a, false, bm, (short)0, acc, false, false);
        }

        const float bias = bvec[nt * 16 + ln];
        float* __restrict__ op =
            out + (size_t)(base_pt + wave * 16) * 32 + nt * 16 + ln;
        #pragma unroll
        for (int j = 0; j < 8; ++j) {
            const int row = j + 8 * lh;      // C/D layout: M = j + 8*(lane>=16)
            op[(size_t)row * 32] = acc[j] + bias;
        }
    }
}

extern "C" void kernel_launch(void* const* d_in, const int* in_sizes, int n_in,
                              void* d_out, int out_size, void* d_ws, size_t ws_size,
                              hipStream_t stream) {
    (void)in_sizes; (void)n_in; (void)out_size; (void)d_ws; (void)ws_size;
    const float* x      = (const float*)d_in[0];
    const float* table  = (const float*)d_in[1];
    const float* shifts = (const float*)d_in[2];
    const float* W      = (const float*)d_in[3];
    const float* b      = (const float*)d_in[4];
    float* out          = (float*)d_out;

    dim3 grid(NPTS / PTS_PER_BLK);
    dim3 block(THREADS);
    permuto_wmma_kernel<<<grid, block, 0, stream>>>(x, table, shifts, W, b, out);
}